// RandomIntervention_64115271794807
// MI455X (gfx1250) — compile-verified
//
#include <hip/hip_runtime.h>
#include <stdint.h>

// CDNA5 / gfx1250: wave32, V_WMMA_F32_16X16X4_F32 (full-fp32 matrix op),
// async global->LDS staging (ASYNCcnt), double-buffered.
typedef __attribute__((ext_vector_type(2))) float v2f;
typedef __attribute__((ext_vector_type(8))) float v8f;

#define FEAT        128
#define NUM_GRAPHS  10000
#define CHUNK       256     // rows per block
#define SUB         64      // rows per staged LDS tile (32 KB packed)
#define NSTAGE      ((SUB * FEAT * 4) / (256 * 16))   // async b128 ops per thread = 8
#define LDS_STRIDE  136     // floats per LDS row; 2*136 % 64 == 16 -> conflict-free half-waves
#define SEG_SENT    (-1073741824)   // INT_MIN/2: one-hot test can never match

__global__ void segsum_zero_kernel(float* __restrict__ out, int n) {
    int i = blockIdx.x * blockDim.x + threadIdx.x;
    if (i < n) out[i] = 0.0f;
}

// Stage SUB rows (addresses clamped to n_nodes-1) into an LDS tile with async
// b128 loads: 256 threads x 8 transfers, each wave streams contiguous 512B.
__device__ __forceinline__ void stage_async(const float* __restrict__ h,
                                            int base_row, int n_nodes,
                                            const float* lds_tile, int tid) {
    unsigned lds_base = (unsigned)(uintptr_t)lds_tile;
#pragma unroll
    for (int i = 0; i < NSTAGE; ++i) {
        int linear = (i * 256 + tid) * 16;           // byte offset in packed 32KB tile
        int rin    = linear >> 9;                    // row within tile (512B rows)
        int row    = min(base_row + rin, n_nodes - 1);
        const char* gp = (const char*)h + ((size_t)row << 9) + (linear & 511);
        unsigned la = lds_base + (unsigned)(rin * (LDS_STRIDE * 4) + (linear & 511));
        asm volatile("global_load_async_to_lds_b128 %0, %1, off th:TH_LOAD_NT"
                     :: "v"(la), "v"(gp) : "memory");
    }
}

__global__ __launch_bounds__(256)
void segsum_wmma_async_kernel(const float* __restrict__ h,
                              const int* __restrict__ batch,
                              float* __restrict__ out,
                              int n_nodes) {
    __shared__ float sh[2][SUB * LDS_STRIDE];   // ~68 KB double buffer
    __shared__ int   sbatch[CHUNK];

    const int tid  = threadIdx.x;
    const int lane = tid & 31;
    const int col  = lane & 15;                 // N index (B/C/D), M index for A
    const int kp   = (lane >> 4) << 1;          // K pair base: 0 or 2
    const int mofs = (lane >> 4) << 3;          // C/D row offset: 0 or 8
    const int feat_base = (tid >> 5) * 16;      // wave -> 16-feature tile

    const int row0 = blockIdx.x * CHUNK;
    if (row0 >= n_nodes) return;
    const int rows_total = min(CHUNK, n_nodes - row0);
    const int nq = (rows_total + SUB - 1) / SUB;

    // Segment ids: valid rows get batch[], tail rows get a sentinel so the
    // one-hot A test fails unconditionally (no per-iteration validity checks).
    sbatch[tid] = (row0 + tid < n_nodes) ? batch[row0 + tid] : SEG_SENT;
    stage_async(h, row0, n_nodes, &sh[0][0], tid);

    v8f acc = {};                               // 16 segs x 16 feats fp32 tile
    // Start far below any real segment: first group does a no-op flush + rebase.
    int base_seg = SEG_SENT / 2;

    auto flush = [&](int base) {
#pragma unroll
        for (int j = 0; j < 8; ++j) {
            float v = acc[j];
            if (v != 0.0f) {
                int seg = base + j + mofs;
                if (seg < NUM_GRAPHS)
                    atomicAdd(&out[(size_t)seg * FEAT + feat_base + col], v);
            }
        }
        acc = (v8f){};
    };

    for (int q = 0; q < nq; ++q) {
        const int buf = q & 1;
        if (q + 1 < nq) {
            stage_async(h, row0 + (q + 1) * SUB, n_nodes, &sh[buf ^ 1][0], tid);
            // Leave exactly the next tile's 8 per-wave async loads outstanding.
            asm volatile("s_wait_asynccnt 0x8" ::: "memory");
        } else {
            asm volatile("s_wait_asynccnt 0x0" ::: "memory");
        }
        __syncthreads();                         // tile q (and sbatch) visible to all waves

        const float* tile = &sh[buf][0];
        const int cl0 = q * SUB;
        const int sub_rows = min(SUB, rows_total - cl0);

        for (int r = 0; r < sub_rows; r += 4) {
            const int cl      = cl0 + r;
            const int s_first = sbatch[cl];                           // valid (cl < rows_total)
            const int s_last  = sbatch[min(cl + 3, rows_total - 1)];  // sorted => group max

            if (s_last - base_seg >= 16) {       // wave-uniform
                flush(base_seg);
                base_seg = s_first;
                if (s_last - base_seg >= 16) {
                    // >=16 empty graphs inside 4 rows: direct atomics from LDS.
#pragma unroll
                    for (int k = 0; k < 2; ++k) {
                        int rr = r + kp + k;
                        int c2 = cl0 + rr;
                        if (c2 < rows_total)
                            atomicAdd(&out[(size_t)sbatch[c2] * FEAT + feat_base + col],
                                      tile[rr * LDS_STRIDE + feat_base + col]);
                    }
                    continue;
                }
            }

            // Unconditional LDS reads. c0 is even -> single aligned ds_load_b64
            // for the lane's K-pair of segment ids; B pair -> ds_load_2addr_b32.
            const int rk0 = r + kp;
            const int c0  = cl + kp;
            const int2 sk = *(const int2*)&sbatch[c0];
            v2f a, b;
            a.x = (sk.x - base_seg == col) ? 1.0f : 0.0f;   // sentinel never matches
            a.y = (sk.y - base_seg == col) ? 1.0f : 0.0f;
            b.x = tile[rk0 * LDS_STRIDE + feat_base + col];
            b.y = tile[(rk0 + 1) * LDS_STRIDE + feat_base + col];

            acc = __builtin_amdgcn_wmma_f32_16x16x4_f32(
                      false, a, false, b, (short)0, acc, false, false);
        }
        __syncthreads();                         // protect buf from next stage's overwrite
    }

    flush(base_seg);
}

extern "C" void kernel_launch(void* const* d_in, const int* in_sizes, int n_in,
                              void* d_out, int out_size, void* d_ws, size_t ws_size,
                              hipStream_t stream) {
    const float* h     = (const float*)d_in[0];   // h_t  : [N_NODES, 128] fp32
    const int*   batch = (const int*)d_in[1];     // batch: [N_NODES] sorted segment ids
    float*       out   = (float*)d_out;           // [NUM_GRAPHS, 128] fp32
    const int n_nodes  = in_sizes[1];

    {   // Harness poisons d_out: zero the atomic-accumulate target first.
        int t = 256, b = (out_size + t - 1) / t;
        segsum_zero_kernel<<<b, t, 0, stream>>>(out, out_size);
    }

    int blocks = (n_nodes + CHUNK - 1) / CHUNK;
    segsum_wmma_async_kernel<<<blocks, 256, 0, stream>>>(h, batch, out, n_nodes);
}